// GPEncoder_40166534152823
// MI455X (gfx1250) — compile-verified
//
#include <hip/hip_runtime.h>

typedef __attribute__((ext_vector_type(2))) float v2f;
typedef __attribute__((ext_vector_type(8))) float v8f;

#define HH 384
#define WW 384
#define HWXY (HH * WW)
#define NB 8
#define ICn 3
#define OCP 32            // output channels padded 30 -> 32
#define KP 28             // im2col K padded 27 -> 28
#define TPB 128           // 4 waves of 32
#define PIX_PER_WG 128
#define TILE_COLS 130     // x0-1 .. x0+128
#define TILE_COLS_P 132   // padded row

__global__ __launch_bounds__(TPB) void gpenc_fused(
    const float* __restrict__ x, const float* __restrict__ Wt,
    const float* __restrict__ bias, float* __restrict__ out)
{
    // LDS staging: input halo tile, weights in B-matrix layout, bias, conv-out transpose buffer
    __shared__ float lds_x[ICn][3][TILE_COLS_P];   // [ic][ky][col]
    __shared__ float lds_w[KP][OCP];               // [k][n] = W[n][k], zero padded
    __shared__ float lds_b[OCP];
    __shared__ float lds_p[4][32][33];             // [wave][pixel][channel] (+1 pad)

    const int tid = threadIdx.x;
    const int wg  = blockIdx.x;
    const int tiles_per_row = WW / PIX_PER_WG;     // 3
    const int bimg = wg / (HH * tiles_per_row);
    const int rem  = wg % (HH * tiles_per_row);
    const int yrow = rem / tiles_per_row;
    const int x0   = (rem % tiles_per_row) * PIX_PER_WG;

    // ---- stage weights (K-major, zero-padded) and bias ----
    for (int i = tid; i < KP * OCP; i += TPB) {
        int k = i >> 5, n = i & 31;
        float v = 0.f;
        if (k < 27 && n < 30) v = Wt[n * 27 + k];  // W[n][ic][ky][kx], k = ic*9+ky*3+kx
        lds_w[k][n] = v;
    }
    if (tid < OCP) lds_b[tid] = (tid < 30) ? bias[tid] : 0.f;

    // ---- stage input halo tile: rows y-1..y+1, cols x0-1..x0+128, zero at borders ----
    for (int i = tid; i < ICn * 3 * TILE_COLS; i += TPB) {
        int ic  = i / (3 * TILE_COLS);
        int r2  = i % (3 * TILE_COLS);
        int ry  = r2 / TILE_COLS;
        int col = r2 % TILE_COLS;
        int gy = yrow + ry - 1;
        int gx = x0 + col - 1;
        float v = 0.f;
        if ((unsigned)gy < (unsigned)HH && (unsigned)gx < (unsigned)WW)
            v = x[(((size_t)bimg * ICn + ic) * HH + gy) * WW + gx];
        lds_x[ic][ry][col] = v;
    }
    __syncthreads();

    // ---- im2col GEMM on the matrix pipe: 2 M-tiles x 2 N-tiles, K = 7 chunks of 4 ----
    const int wv   = tid >> 5;
    const int lane = tid & 31;
    const int half = lane >> 4;        // K sub-pair select per ISA A/B f32 layout
    const int n    = lane & 15;        // N (channel) / M (pixel-in-tile) index
    const int px0  = wv * 32 + n;      // local pixel of M-tile 0
    const int px1  = px0 + 16;         // local pixel of M-tile 1

    v8f acc00 = {}, acc01 = {}, acc10 = {}, acc11 = {};
#pragma unroll
    for (int c = 0; c < 7; ++c) {
        const int kb = 4 * c + 2 * half;
        v2f a0, a1, b0, b1;
#pragma unroll
        for (int e = 0; e < 2; ++e) {
            int k  = kb + e;
            int ks = (k < 27) ? k : 0;         // safe decode for padded K
            int ic = ks / 9;
            int r9 = ks - 9 * ic;
            int ky = r9 / 3;
            int kx = r9 - 3 * ky;
            float va0 = lds_x[ic][ky][px0 + kx];
            float va1 = lds_x[ic][ky][px1 + kx];
            a0[e] = (k < 27) ? va0 : 0.f;
            a1[e] = (k < 27) ? va1 : 0.f;
            b0[e] = lds_w[k][n];               // row 27 is zero-padded
            b1[e] = lds_w[k][n + 16];
        }
        acc00 = __builtin_amdgcn_wmma_f32_16x16x4_f32(false, a0, false, b0, (short)0, acc00, false, false);
        acc01 = __builtin_amdgcn_wmma_f32_16x16x4_f32(false, a0, false, b1, (short)0, acc01, false, false);
        acc10 = __builtin_amdgcn_wmma_f32_16x16x4_f32(false, a1, false, b0, (short)0, acc10, false, false);
        acc11 = __builtin_amdgcn_wmma_f32_16x16x4_f32(false, a1, false, b1, (short)0, acc11, false, false);
    }

    // ---- transpose accumulators through LDS: p[pixel][channel] ----
    // C/D layout: lanes 0-15 -> N=lane, VGPR r -> M=r; lanes 16-31 -> N=lane-16, M=r+8
#pragma unroll
    for (int r = 0; r < 8; ++r) {
        int pr = r + 8 * half;
        lds_p[wv][pr][n]            = acc00[r];
        lds_p[wv][pr][n + 16]       = acc01[r];
        lds_p[wv][16 + pr][n]       = acc10[r];
        lds_p[wv][16 + pr][n + 16]  = acc11[r];
    }
    __syncthreads();

    // ---- per-pixel tail: one lane = one pixel (tid = wv*32 + lane) ----
    float p[30];
#pragma unroll
    for (int ch = 0; ch < 30; ++ch)
        p[ch] = lds_p[wv][lane][ch] + lds_b[ch];

    const int   gx  = x0 + tid;
    const size_t off = (size_t)yrow * WW + gx;
    const float TWO_PI = 6.28318530717958647692f;

    float mv[6], sv[6], xl[3], angv[15];
#pragma unroll
    for (int i = 0; i < 6; ++i) mv[i] = p[i];
#pragma unroll
    for (int i = 0; i < 6; ++i) { float t = p[6 + i]; sv[i] = t * t; }  // relu(x)-relu(-x) == x
#pragma unroll
    for (int i = 0; i < 15; ++i) {
        float t = p[12 + i];
        angv[i] = TWO_PI / (1.f + __expf(-t));  // 2*pi*sigmoid
    }
#pragma unroll
    for (int i = 0; i < 3; ++i) xl[i] = p[27 + i];

    // hyperspherical Cholesky: L[i][j] = cos(th_ij) * prod_{k<j} sin(th_ik); L[i][i] = prod sin
    float L[6][6];
    {
        int t = 0;  // tril_indices(6,-1) row-major order
#pragma unroll
        for (int i = 0; i < 6; ++i) {
            float prod = 1.f;
#pragma unroll
            for (int j = 0; j < i; ++j) {
                float th = angv[t++];
                L[i][j] = __cosf(th) * prod;
                prod *= __sinf(th);
            }
            L[i][i] = prod;
        }
    }

    // r = L * L^T (symmetric, only k <= min(i,j) terms nonzero)
    float rm[6][6];
#pragma unroll
    for (int i = 0; i < 6; ++i)
#pragma unroll
        for (int j = 0; j <= i; ++j) {
            float acc = 0.f;
#pragma unroll
            for (int k2 = 0; k2 <= j; ++k2) acc += L[i][k2] * L[j][k2];
            rm[i][j] = acc;
            rm[j][i] = acc;
        }

    // z[c] = s*(y0-m); y0[3:6]==m[3:6] so z[3:6]==0 exactly -> only c=0..2 contribute
    float z[3];
#pragma unroll
    for (int i = 0; i < 3; ++i) z[i] = sv[i] * (xl[i] - mv[i]);
    float yv[3];
#pragma unroll
    for (int l = 0; l < 3; ++l) {
        float acc = z[0] * rm[0][3 + l] + z[1] * rm[1][3 + l] + z[2] * rm[2][3 + l];
        yv[l] = mv[3 + l] + acc / sv[3 + l];
    }

    // ---- coalesced plane writes (128 contiguous floats per plane per block) ----
    const size_t m_base = (size_t)NB * 3 * HWXY;                 // after xl
    const size_t c_base = m_base + (size_t)NB * 6 * HWXY;        // after m
    const size_t y_base = c_base + (size_t)NB * 36 * HWXY;       // after c

#pragma unroll
    for (int i = 0; i < 3; ++i)
        out[((size_t)bimg * 3 + i) * HWXY + off] = xl[i];
#pragma unroll
    for (int i = 0; i < 6; ++i)
        out[m_base + ((size_t)bimg * 6 + i) * HWXY + off] = mv[i];
#pragma unroll
    for (int i = 0; i < 6; ++i)
#pragma unroll
        for (int j = 0; j < 6; ++j)
            out[c_base + ((size_t)bimg * 36 + i * 6 + j) * HWXY + off] = rm[i][j];
#pragma unroll
    for (int k = 0; k < 3; ++k)
        out[y_base + ((size_t)bimg * 3 + k) * HWXY + off] = yv[k];
}

extern "C" void kernel_launch(void* const* d_in, const int* in_sizes, int n_in,
                              void* d_out, int out_size, void* d_ws, size_t ws_size,
                              hipStream_t stream) {
    (void)in_sizes; (void)n_in; (void)out_size; (void)d_ws; (void)ws_size;
    const float* x = (const float*)d_in[0];
    const float* W = (const float*)d_in[1];
    const float* b = (const float*)d_in[2];
    float* out = (float*)d_out;

    const int blocks = NB * HH * (WW / PIX_PER_WG);  // 8 * 384 * 3 = 9216
    gpenc_fused<<<blocks, TPB, 0, stream>>>(x, W, b, out);
}